// Projection_52175262711987
// MI455X (gfx1250) — compile-verified
//
#include <hip/hip_runtime.h>
#include <hip/hip_bf16.h>
#include <stdint.h>

#define IMG_H 480
#define IMG_W 640
#define IMG_HW (IMG_H * IMG_W)
#define N_PTS 4000000

typedef __attribute__((ext_vector_type(2))) float v2f;
typedef __attribute__((ext_vector_type(8))) float v8f;

// ---------------------------------------------------------------------------
// Kernel 1: fused prep.
//  - every thread clears 16B of the u64 z-buffer (min-identity = all ones)
//  - block 0 / thread 0 additionally computes alpha = max|x|,
//    ext = E @ Tz(alpha), 4x4 adjugate inverse, P = K @ inv_ext[0:3,:]
// ---------------------------------------------------------------------------
__global__ void prep_kernel(const float* __restrict__ x,
                            const float* __restrict__ extr,
                            const float* __restrict__ intr,
                            float* __restrict__ P,
                            ulonglong2* __restrict__ zb2) {
    int i = blockIdx.x * blockDim.x + threadIdx.x;
    if (i < IMG_HW / 2) {
        zb2[i] = make_ulonglong2(0xFFFFFFFFFFFFFFFFull, 0xFFFFFFFFFFFFFFFFull);
    }
    if (threadIdx.x != 0 || blockIdx.x != 0) return;

    float alpha = 0.0f;
    #pragma unroll
    for (int k = 0; k < 16; ++k) alpha = fmaxf(alpha, fabsf(x[k]));

    // ext = extr @ Tz(alpha): last column += alpha * third column
    float m[16];
    #pragma unroll
    for (int k = 0; k < 16; ++k) m[k] = extr[k];
    #pragma unroll
    for (int r = 0; r < 4; ++r) m[r*4+3] += alpha * m[r*4+2];

    // 4x4 inverse via adjugate (row-major)
    float iv[16];
    iv[0]  =  m[5]*m[10]*m[15] - m[5]*m[11]*m[14] - m[9]*m[6]*m[15] + m[9]*m[7]*m[14] + m[13]*m[6]*m[11] - m[13]*m[7]*m[10];
    iv[4]  = -m[4]*m[10]*m[15] + m[4]*m[11]*m[14] + m[8]*m[6]*m[15] - m[8]*m[7]*m[14] - m[12]*m[6]*m[11] + m[12]*m[7]*m[10];
    iv[8]  =  m[4]*m[9]*m[15]  - m[4]*m[11]*m[13] - m[8]*m[5]*m[15] + m[8]*m[7]*m[13] + m[12]*m[5]*m[11] - m[12]*m[7]*m[9];
    iv[12] = -m[4]*m[9]*m[14]  + m[4]*m[10]*m[13] + m[8]*m[5]*m[14] - m[8]*m[6]*m[13] - m[12]*m[5]*m[10] + m[12]*m[6]*m[9];
    iv[1]  = -m[1]*m[10]*m[15] + m[1]*m[11]*m[14] + m[9]*m[2]*m[15] - m[9]*m[3]*m[14] - m[13]*m[2]*m[11] + m[13]*m[3]*m[10];
    iv[5]  =  m[0]*m[10]*m[15] - m[0]*m[11]*m[14] - m[8]*m[2]*m[15] + m[8]*m[3]*m[14] + m[12]*m[2]*m[11] - m[12]*m[3]*m[10];
    iv[9]  = -m[0]*m[9]*m[15]  + m[0]*m[11]*m[13] + m[8]*m[1]*m[15] - m[8]*m[3]*m[13] - m[12]*m[1]*m[11] + m[12]*m[3]*m[9];
    iv[13] =  m[0]*m[9]*m[14]  - m[0]*m[10]*m[13] - m[8]*m[1]*m[14] + m[8]*m[2]*m[13] + m[12]*m[1]*m[10] - m[12]*m[2]*m[9];
    iv[2]  =  m[1]*m[6]*m[15]  - m[1]*m[7]*m[14]  - m[5]*m[2]*m[15] + m[5]*m[3]*m[14] + m[13]*m[2]*m[7]  - m[13]*m[3]*m[6];
    iv[6]  = -m[0]*m[6]*m[15]  + m[0]*m[7]*m[14]  + m[4]*m[2]*m[15] - m[4]*m[3]*m[14] - m[12]*m[2]*m[7]  + m[12]*m[3]*m[6];
    iv[10] =  m[0]*m[5]*m[15]  - m[0]*m[7]*m[13]  - m[4]*m[1]*m[15] + m[4]*m[3]*m[13] + m[12]*m[1]*m[7]  - m[12]*m[3]*m[5];
    iv[14] = -m[0]*m[5]*m[14]  + m[0]*m[6]*m[13]  + m[4]*m[1]*m[14] - m[4]*m[2]*m[13] - m[12]*m[1]*m[6]  + m[12]*m[2]*m[5];
    iv[3]  = -m[1]*m[6]*m[11]  + m[1]*m[7]*m[10]  + m[5]*m[2]*m[11] - m[5]*m[3]*m[10] - m[9]*m[2]*m[7]   + m[9]*m[3]*m[6];
    iv[7]  =  m[0]*m[6]*m[11]  - m[0]*m[7]*m[10]  - m[4]*m[2]*m[11] + m[4]*m[3]*m[10] + m[8]*m[2]*m[7]   - m[8]*m[3]*m[6];
    iv[11] = -m[0]*m[5]*m[11]  + m[0]*m[7]*m[9]   + m[4]*m[1]*m[11] - m[4]*m[3]*m[9]  - m[8]*m[1]*m[7]   + m[8]*m[3]*m[5];
    iv[15] =  m[0]*m[5]*m[10]  - m[0]*m[6]*m[9]   - m[4]*m[1]*m[10] + m[4]*m[2]*m[9]  + m[8]*m[1]*m[6]   - m[8]*m[2]*m[5];
    float det = m[0]*iv[0] + m[1]*iv[4] + m[2]*iv[8] + m[3]*iv[12];
    float rd = 1.0f / det;

    // P(3x4) = K(3x3) @ inv_ext(3x4)
    #pragma unroll
    for (int r = 0; r < 3; ++r) {
        #pragma unroll
        for (int c = 0; c < 4; ++c) {
            float s = 0.0f;
            #pragma unroll
            for (int k = 0; k < 3; ++k) s += intr[r*3+k] * (iv[k*4+c] * rd);
            P[r*4+c] = s;
        }
    }
}

// ---------------------------------------------------------------------------
// Kernel 2: WMMA projection + packed-key atomicMin z-buffer scatter.
// Branch-free software pipeline: the next chunk's (clamped-index) load is
// issued unconditionally at the top of the body; its first use is the loop
// latch, so the loadcnt wait sinks past the WMMAs and the atomic.
// ---------------------------------------------------------------------------
#define SCAT_BLOCK 256
#define WAVE_TILE  (32 * 17)   // 32 rows, stride 17 floats (bank-conflict pad)

__global__ __launch_bounds__(SCAT_BLOCK)
void project_scatter_kernel(const float* __restrict__ pc,
                            const float* __restrict__ P,
                            unsigned long long* __restrict__ zb) {
    __shared__ float sbuf[(SCAT_BLOCK / 32) * WAVE_TILE];

    const int lane      = threadIdx.x & 31;
    const int waveInBlk = threadIdx.x >> 5;
    float* ws = &sbuf[waveInBlk * WAVE_TILE];

    const int gwave   = (int)((blockIdx.x * blockDim.x + threadIdx.x) >> 5);
    const int nwaves  = (int)((gridDim.x * blockDim.x) >> 5);
    const int nChunks = N_PTS / 32;

    // B = P^T (4x16), wave-constant. f32 B layout (mirrors documented A 16x4):
    // lane n (0-15): column n, K=0 in v0, K=1 in v1; lane n+16: K=2,3.
    const int nn = lane & 15;
    const int kh = lane >> 4;
    float b0 = 0.0f, b1 = 0.0f;
    if (nn < 3) { b0 = P[nn*4 + 2*kh]; b1 = P[nn*4 + 2*kh + 1]; }
    v2f B; B[0] = b0; B[1] = b1;
    v8f Cz = {};

    // Prologue: load first chunk's coordinates (guarded by c < nChunks).
    int c = gwave;
    {
        int c0 = (c < nChunks) ? c : 0;
        const float* q = pc + (size_t)(c0 * 32 + lane) * 6;
        // fallthrough values only consumed if c < nChunks
    }
    float x = 0.0f, y = 0.0f, z = 1.0f;
    if (c < nChunks) {                       // wave-uniform; executes at most once
        const float* q = pc + (size_t)(c * 32 + lane) * 6;
        x = q[0]; y = q[1]; z = q[2];
    }

    for (; c < nChunks; c += nwaves) {       // wave-uniform: EXEC all-1s at WMMA
        // Unconditional clamped prefetch of the next chunk: no divergent
        // branch, so the wait for this load can sink to the loop latch.
        int cn = c + nwaves;
        int cl = (cn < nChunks) ? cn : (nChunks - 1);
        const float* qn = pc + (size_t)(cl * 32 + lane) * 6;
        float xn = qn[0];
        float yn = qn[1];
        float zn = qn[2];

        // A1 = points c*32 .. +15 (16x4):
        //   lanes 0-15: row L -> (x_L, y_L); lanes 16-31: row L-16 -> (z, 1)
        float zlo = __shfl(z, lane & 15, 32);
        v2f A1; A1[0] = (lane < 16) ? x : zlo;  A1[1] = (lane < 16) ? y : 1.0f;
        // A2 = points c*32+16 .. +31
        float xhi = __shfl(x, (lane & 15) + 16, 32);
        float yhi = __shfl(y, (lane & 15) + 16, 32);
        v2f A2; A2[0] = (lane < 16) ? xhi : z;  A2[1] = (lane < 16) ? yhi : 1.0f;

        // D[m][n] = proj component n of point m
        v8f d1 = __builtin_amdgcn_wmma_f32_16x16x4_f32(false, A1, false, B, (short)0, Cz, false, false);
        v8f d2 = __builtin_amdgcn_wmma_f32_16x16x4_f32(false, A2, false, B, (short)0, Cz, false, false);

        // Transpose D back: row m = chunk-local point, col n = component
        const int colb = lane & 15;
        const int mofs = (lane >> 4) * 8;
        #pragma unroll
        for (int r = 0; r < 8; ++r) {
            ws[(mofs + r) * 17 + colb]      = d1[r];
            ws[(16 + mofs + r) * 17 + colb] = d2[r];
        }
        asm volatile("s_wait_dscnt 0" ::: "memory");   // same-wave LDS ordering + fence

        float pjx = ws[lane * 17 + 0];
        float pjy = ws[lane * 17 + 1];
        float dep = ws[lane * 17 + 2];

        // One reciprocal (+1 Newton step) replaces two IEEE divides.
        float invd = __builtin_amdgcn_rcpf(dep);
        invd = invd * fmaf(-dep, invd, 2.0f);
        float u = pjx * invd;
        float v = pjy * invd;

        int px = (int)floorf(u);
        int py = (int)floorf(v);
        bool valid = (px >= 0) & (py >= 0) & (px < IMG_W) & (py < IMG_H) & (dep > 0.0f);
        if (valid) {
            int pix = py * IMG_W + px;
            // depth>0 => float bits order-preserving; low 32 bits = index tie-break
            unsigned long long key =
                ((unsigned long long)__float_as_uint(dep) << 32) |
                (unsigned)(c * 32 + lane);
            atomicMin(&zb[pix], key);
        }
        asm volatile("" ::: "memory");   // keep next iter's LDS stores after these loads

        x = xn; y = yn; z = zn;          // first use of prefetch: loop latch
    }
}

// ---------------------------------------------------------------------------
// Kernel 3: resolve winners -> (1,3,H,W), background 1.0; 2 pixels/thread
// ---------------------------------------------------------------------------
__global__ void resolve_kernel(const ulonglong2* __restrict__ zb2,
                               const float* __restrict__ pc,
                               float* __restrict__ out) {
    int i = blockIdx.x * blockDim.x + threadIdx.x;   // pixel pair index
    if (i >= IMG_HW / 2) return;
    ulonglong2 kk = zb2[i];

    float r0 = 1.0f, g0 = 1.0f, b0 = 1.0f;
    float r1 = 1.0f, g1 = 1.0f, b1 = 1.0f;
    if (kk.x != 0xFFFFFFFFFFFFFFFFull) {
        const float* q = pc + (size_t)((unsigned)kk.x) * 6;
        r0 = q[3]; g0 = q[4]; b0 = q[5];
    }
    if (kk.y != 0xFFFFFFFFFFFFFFFFull) {
        const float* q = pc + (size_t)((unsigned)kk.y) * 6;
        r1 = q[3]; g1 = q[4]; b1 = q[5];
    }
    float2* o0 = (float2*)(out)             + i;
    float2* o1 = (float2*)(out + IMG_HW)    + i;
    float2* o2 = (float2*)(out + 2*IMG_HW)  + i;
    *o0 = make_float2(r0, r1);
    *o1 = make_float2(g0, g1);
    *o2 = make_float2(b0, b1);
}

// ---------------------------------------------------------------------------
extern "C" void kernel_launch(void* const* d_in, const int* in_sizes, int n_in,
                              void* d_out, int out_size, void* d_ws, size_t ws_size,
                              hipStream_t stream) {
    const float* pc   = (const float*)d_in[0];  // (4M, 6)
    const float* x    = (const float*)d_in[1];  // (1, 16)
    const float* extr = (const float*)d_in[2];  // (4, 4)
    const float* intr = (const float*)d_in[3];  // (3, 3)
    float* out = (float*)d_out;                 // (1, 3, 480, 640)

    float* P = (float*)d_ws;                                            // 12 floats
    unsigned long long* zb = (unsigned long long*)((char*)d_ws + 256);  // HW u64

    prep_kernel<<<(IMG_HW / 2 + 255) / 256, 256, 0, stream>>>(
        x, extr, intr, P, (ulonglong2*)zb);
    project_scatter_kernel<<<1024, SCAT_BLOCK, 0, stream>>>(pc, P, zb);
    resolve_kernel<<<(IMG_HW / 2 + 255) / 256, 256, 0, stream>>>(
        (const ulonglong2*)zb, pc, out);
}